// LocalInteractionsLayer_16518444220433
// MI455X (gfx1250) — compile-verified
//
#include <hip/hip_runtime.h>

// CDNA5 / gfx1250: one wave32 per (b, c, p); V_WMMA_F32_16X16X4_F32 computes
// the 15x15 (padded 16x16) outer product top_w ⊗ bot_w exactly in f32.

typedef __attribute__((ext_vector_type(2))) float v2f;
typedef __attribute__((ext_vector_type(8))) float v8f;

namespace {
constexpr int kB = 16;
constexpr int kC = 8;
constexpr int kL = 4096;
constexpr int kW = 15;                 // 2*7+1
constexpr int kP = kL - 14;            // 4082
constexpr int kOutPerC = kW * kW;      // 225
constexpr int kOutPerBP = kC * kOutPerC; // 1800
}

__global__ __launch_bounds__(256)
void local_interactions_wmma(const float* __restrict__ seq, float* __restrict__ out) {
  const int lane = threadIdx.x & 31;
  const int waveInBlk = threadIdx.x >> 5;
  const long long wid = (long long)blockIdx.x * (blockDim.x >> 5) + waveInBlk;
  const long long total = (long long)kB * kC * kP;
  if (wid >= total) return;            // wave-uniform: EXEC stays all-1s

  const int p = (int)(wid % kP);
  const int t = (int)(wid / kP);
  const int c = t % kC;
  const int b = t / kC;

  // Each lane loads the interleaved (top, bot) pair at window offset m = lane&15.
  const int m = lane & 15;
  const long long row = (long long)(b * kC + c) * kL;
  long long lidx = row + p + m;
  const long long lmax = row + (kL - 1);
  if (lidx > lmax) lidx = lmax;        // m==15 slot is unused; clamp to stay in-bounds
  const float2 tb = *(const float2*)(seq + lidx * 2);

  // A (16x4 f32): documented layout -> VGPR0 lanes 0-15 hold K=0, M=lane.
  // Only K=0 slice is non-zero: top values for M=0..14.
  v2f a;
  a.x = (lane < 15) ? tb.x : 0.0f;     // lanes 16-31 (K=2) and M=15 pad -> 0
  a.y = 0.0f;                          // K=1,3 -> 0

  // B (4x16 f32): replicate bot[n] into every K row (layout-proof); only A's
  // K=0 is non-zero, so D[m][n] = top[m] * bot[n].
  const float bv = (m < 15) ? tb.y : 0.0f;
  v2f bm;
  bm.x = bv;
  bm.y = bv;

  v8f cacc = {};
  // 8 args: (neg_a, A, neg_b, B, c_mod, C, reuse_a, reuse_b)
  v8f d = __builtin_amdgcn_wmma_f32_16x16x4_f32(
      /*neg_a=*/false, a, /*neg_b=*/false, bm,
      /*c_mod=*/(short)0, cacc, /*reuse_a=*/false, /*reuse_b=*/false);

  // D layout: VGPR v, lanes 0-15 -> M=v, N=lane; lanes 16-31 -> M=v+8, N=lane-16.
  // out[b, p, c*225 + M*15 + N], contiguous 225-float block per wave.
  const size_t base = ((size_t)b * kP + p) * (size_t)kOutPerBP + (size_t)c * kOutPerC;
  const int M0 = (lane >> 4) * 8;
  const int N = m;

  if (N < kW) {
#pragma unroll
    for (int v = 0; v < 8; ++v) {
      const int M = M0 + v;
      if (M < kW) {
        __builtin_nontemporal_store(d[v], out + base + (size_t)M * kW + N);
      }
    }
  }
}

extern "C" void kernel_launch(void* const* d_in, const int* in_sizes, int n_in,
                              void* d_out, int out_size, void* d_ws, size_t ws_size,
                              hipStream_t stream) {
  (void)in_sizes; (void)n_in; (void)d_ws; (void)ws_size; (void)out_size;
  const float* seq = (const float*)d_in[0];
  float* out = (float*)d_out;

  const long long totalWaves = (long long)kB * kC * kP;  // 522,496
  const int wavesPerBlock = 8;                            // 256 threads
  const int blocks = (int)((totalWaves + wavesPerBlock - 1) / wavesPerBlock);
  local_interactions_wmma<<<blocks, wavesPerBlock * 32, 0, stream>>>(seq, out);
}